// RGATConv_10496900071977
// MI455X (gfx1250) — compile-verified
//
#include <hip/hip_runtime.h>

// ---------------------------------------------------------------------------
// RGATConv for MI455X (gfx1250, wave32).
// Dense GEMMs via V_WMMA_F32_16X16X4_F32 (exact fp32), edge softmax via
// encoded-float atomic max + atomic denom + wave-per-edge scatter.
// ---------------------------------------------------------------------------

typedef float v2f __attribute__((ext_vector_type(2)));
typedef float v8f __attribute__((ext_vector_type(8)));

#define DD 64   // input feature dim
#define FF 64   // output feature dim
#define NEG_SLOPE 0.2f

__device__ __forceinline__ float leaky(float v) {
    return v > 0.f ? v : NEG_SLOPE * v;
}

// Monotonic float -> uint encoding so atomicMax(uint) == float max.
__device__ __forceinline__ unsigned fenc(float f) {
    int i = __float_as_int(f);
    return (i >= 0) ? ((unsigned)i | 0x80000000u) : ~(unsigned)i;
}
__device__ __forceinline__ float fdec(unsigned u) {
    int i = (u & 0x80000000u) ? (int)(u & 0x7FFFFFFFu) : ~(int)u;
    return __int_as_float(i);
}

// ---------------------------------------------------------------------------
// Kernel 1: H[r] = x @ W[r] for r in [0,R), H[R] = x @ W_self.
// Fused: S[r][n] = h_r[n] . att_src[r], Dt[r][n] = h_r[n] . att_dst[r].
// One block = one (r, 16-node tile); 4 waves each own a 16-column group.
// WMMA f32 16x16x4, K-loop of 16 steps over LDS-staged tiles using the
// documented CDNA5 f32 A/B/C lane layouts.
// ---------------------------------------------------------------------------
__global__ __launch_bounds__(128) void rgat_gemm_kernel(
    const float* __restrict__ x,        // [N,64]
    const float* __restrict__ W,        // [R,64,64]
    const float* __restrict__ W_self,   // [64,64]
    const float* __restrict__ att_src,  // [R,64]
    const float* __restrict__ att_dst,  // [R,64]
    float* __restrict__ H,              // [(R+1),N,64]
    float* __restrict__ S,              // [R,N]
    float* __restrict__ Dt,             // [R,N]
    int N, int R)
{
    __shared__ float Xs[16][DD + 1];    // 16-node tile of x (padded rows)
    __shared__ float Ws[DD][FF];        // full 64x64 weight tile
    __shared__ float sAcc[16];
    __shared__ float dAcc[16];

    const int numTiles = (N + 15) >> 4;
    const int r    = blockIdx.x / numTiles;   // 0..R ; R == self slot
    const int tile = blockIdx.x % numTiles;
    const int node0 = tile << 4;

    const float* Wmat = (r < R) ? (W + (size_t)r * DD * FF) : W_self;

    const int tid = threadIdx.x;

    // Cooperative loads (fully coalesced).
    for (int i = tid; i < DD * FF; i += 128)
        Ws[i >> 6][i & 63] = Wmat[i];
    for (int i = tid; i < 16 * DD; i += 128) {
        int m = i >> 6, k = i & 63;
        int node = node0 + m;
        Xs[m][k] = (node < N) ? x[(size_t)node * DD + k] : 0.f;
    }
    if (tid < 16) { sAcc[tid] = 0.f; dAcc[tid] = 0.f; }
    __syncthreads();

    const int wave = tid >> 5;        // 0..3 -> column group
    const int lane = tid & 31;
    const int hi   = lane >> 4;       // lane half
    const int li   = lane & 15;
    const int n0   = wave << 4;

    // Accumulator: 16x16 f32 C/D tile. VGPR i holds row (i + 8*hi), col li.
    v8f c = {};
    #pragma unroll
    for (int kk = 0; kk < 16; ++kk) {
        const int ka = (kk << 2) + (hi << 1);  // K base for this lane half
        v2f a, b;
        // A 16x4 f32: lane li = row, VGPR0/1 = K ka/ka+1 (halves offset by 2)
        a.x = Xs[li][ka];
        a.y = Xs[li][ka + 1];
        // B 4x16 f32: lane li = col, VGPR0/1 = K ka/ka+1 (same half split)
        b.x = Ws[ka][n0 + li];
        b.y = Ws[ka + 1][n0 + li];
        c = __builtin_amdgcn_wmma_f32_16x16x4_f32(
                false, a, false, b, (short)0, c, false, false);
    }

    // Store H tile + fused attention projections.
    float* Hr = H + (size_t)r * N * FF;
    float as = 0.f, ad = 0.f;
    if (r < R) {
        as = att_src[(size_t)r * FF + n0 + li];
        ad = att_dst[(size_t)r * FF + n0 + li];
    }
    #pragma unroll
    for (int i = 0; i < 8; ++i) {
        const int m = i + (hi << 3);
        const int node = node0 + m;
        if (node < N) {
            Hr[(size_t)node * FF + n0 + li] = c[i];
            if (r < R) {
                atomicAdd(&sAcc[m], c[i] * as);   // LDS ds_add_f32
                atomicAdd(&dAcc[m], c[i] * ad);
            }
        }
    }
    if (r < R) {
        __syncthreads();
        if (tid < 16) {
            const int node = node0 + tid;
            if (node < N) {
                S [(size_t)r * N + node] = sAcc[tid];
                Dt[(size_t)r * N + node] = dAcc[tid];
            }
        }
    }
}

// ---------------------------------------------------------------------------
// Kernel 2: seed segment-max with the (always-present) self-loop term.
// Self-loop for (r,n): e = leaky(S[r][n] + Dt[r][n]).
// ---------------------------------------------------------------------------
__global__ void rgat_selfmax_kernel(const float* __restrict__ S,
                                    const float* __restrict__ Dt,
                                    unsigned* __restrict__ Menc,
                                    int RN)
{
    int i = blockIdx.x * blockDim.x + threadIdx.x;
    if (i < RN) Menc[i] = fenc(leaky(S[i] + Dt[i]));
}

// ---------------------------------------------------------------------------
// Kernel 3: edge pass 1 — atomic max of leaky(S[r][src]+Dt[r][dst]) per dst.
// ---------------------------------------------------------------------------
__global__ void rgat_edgemax_kernel(const int* __restrict__ src,
                                    const int* __restrict__ dst,
                                    const int* __restrict__ etype,
                                    const float* __restrict__ S,
                                    const float* __restrict__ Dt,
                                    unsigned* __restrict__ Menc,
                                    int N, int E)
{
    int e = blockIdx.x * blockDim.x + threadIdx.x;
    if (e >= E) return;
    const int r = etype[e];
    const float ev = leaky(S[(size_t)r * N + src[e]] + Dt[(size_t)r * N + dst[e]]);
    atomicMax(&Menc[(size_t)r * N + dst[e]], fenc(ev));
}

// ---------------------------------------------------------------------------
// Kernel 4: denom init with self-loop exp term.
// ---------------------------------------------------------------------------
__global__ void rgat_deninit_kernel(const float* __restrict__ S,
                                    const float* __restrict__ Dt,
                                    const unsigned* __restrict__ Menc,
                                    float* __restrict__ Den,
                                    int RN)
{
    int i = blockIdx.x * blockDim.x + threadIdx.x;
    if (i < RN) Den[i] = __expf(leaky(S[i] + Dt[i]) - fdec(Menc[i]));
}

// ---------------------------------------------------------------------------
// Kernel 5: edge pass 2 — atomicAdd exp(e - m[dst]) into denom.
// ---------------------------------------------------------------------------
__global__ void rgat_edgeden_kernel(const int* __restrict__ src,
                                    const int* __restrict__ dst,
                                    const int* __restrict__ etype,
                                    const float* __restrict__ S,
                                    const float* __restrict__ Dt,
                                    const unsigned* __restrict__ Menc,
                                    float* __restrict__ Den,
                                    int N, int E)
{
    int e = blockIdx.x * blockDim.x + threadIdx.x;
    if (e >= E) return;
    const int r = etype[e];
    const size_t dIdx = (size_t)r * N + dst[e];
    const float ev = leaky(S[(size_t)r * N + src[e]] + Dt[dIdx]);
    atomicAdd(&Den[dIdx], __expf(ev - fdec(Menc[dIdx])));
}

// ---------------------------------------------------------------------------
// Kernel 6: deterministic base write of out:
//   out[n][f] = Hself[n][f] + b_self[f]
//             + sum_r ( bias[r][f] + alpha_self(r,n) * H[r][n][f] )
// Must run before the edge scatter (plain store, then atomics accumulate).
// ---------------------------------------------------------------------------
__global__ void rgat_baseout_kernel(const float* __restrict__ H,   // [(R+1),N,64]
                                    const float* __restrict__ S,
                                    const float* __restrict__ Dt,
                                    const unsigned* __restrict__ Menc,
                                    const float* __restrict__ Den,
                                    const float* __restrict__ b_self,
                                    const float* __restrict__ bias, // [R,64]
                                    float* __restrict__ out,
                                    int N, int R)
{
    const size_t i = (size_t)blockIdx.x * blockDim.x + threadIdx.x;
    if (i >= (size_t)N * FF) return;
    const int n = (int)(i >> 6);
    const int f = (int)(i & 63);

    float acc = H[(size_t)R * N * FF + i] + b_self[f];  // self GEMM slot
    for (int r = 0; r < R; ++r) {
        const size_t idx = (size_t)r * N + n;
        const float aself = __expf(leaky(S[idx] + Dt[idx]) - fdec(Menc[idx])) / Den[idx];
        acc += bias[(size_t)r * FF + f] + aself * H[(size_t)r * N * FF + i];
    }
    out[i] = acc;
}

// ---------------------------------------------------------------------------
// Kernel 7: edge pass 3 — one wave32 per edge. Lane l handles features
// l and l+32: coalesced 128B gather of h[src] row, atomic scatter into out.
// ---------------------------------------------------------------------------
__global__ __launch_bounds__(256) void rgat_scatter_kernel(
    const int* __restrict__ src,
    const int* __restrict__ dst,
    const int* __restrict__ etype,
    const float* __restrict__ S,
    const float* __restrict__ Dt,
    const unsigned* __restrict__ Menc,
    const float* __restrict__ Den,
    const float* __restrict__ H,
    float* __restrict__ out,
    int N, int E)
{
    const int gtid = blockIdx.x * blockDim.x + threadIdx.x;
    const int e    = gtid >> 5;          // one wave32 per edge
    const int lane = threadIdx.x & 31;
    if (e >= E) return;

    const int r = etype[e];
    const int s = src[e];
    const int d = dst[e];
    const size_t dIdx = (size_t)r * N + d;
    // All lanes recompute the scalar alpha; loads are uniform -> broadcast.
    const float ev    = leaky(S[(size_t)r * N + s] + Dt[dIdx]);
    const float alpha = __expf(ev - fdec(Menc[dIdx])) / Den[dIdx];

    const float* hs = H + ((size_t)r * N + s) * FF;
    float* op = out + (size_t)d * FF;
    atomicAdd(&op[lane],      alpha * hs[lane]);
    atomicAdd(&op[lane + 32], alpha * hs[lane + 32]);
}

// ---------------------------------------------------------------------------
// Host-side launcher (graph-capture safe: only kernel launches on `stream`).
// ---------------------------------------------------------------------------
extern "C" void kernel_launch(void* const* d_in, const int* in_sizes, int n_in,
                              void* d_out, int out_size, void* d_ws, size_t ws_size,
                              hipStream_t stream)
{
    const float* x       = (const float*)d_in[0];
    const int*   ei      = (const int*)  d_in[1];   // [2,E] flat: src then dst
    const int*   etype   = (const int*)  d_in[2];
    const float* W_self  = (const float*)d_in[3];
    const float* b_self  = (const float*)d_in[4];
    const float* W       = (const float*)d_in[5];
    const float* att_src = (const float*)d_in[6];
    const float* att_dst = (const float*)d_in[7];
    const float* bias    = (const float*)d_in[8];
    float* out = (float*)d_out;

    const int N = in_sizes[0] / DD;
    const int E = in_sizes[1] / 2;
    const int R = in_sizes[5] / (DD * FF);

    const int* src = ei;
    const int* dst = ei + E;

    // Workspace carve-up (floats):
    //   H   : (R+1)*N*64   (slot R = self GEMM)
    //   S,Dt: R*N each
    //   Menc: R*N (uint)   Den: R*N
    float* ws = (float*)d_ws;
    size_t off = 0;
    float*    H    = ws + off; off += (size_t)(R + 1) * N * FF;
    float*    S    = ws + off; off += (size_t)R * N;
    float*    Dt   = ws + off; off += (size_t)R * N;
    unsigned* Menc = (unsigned*)(ws + off); off += (size_t)R * N;
    float*    Den  = ws + off; off += (size_t)R * N;

    const int numTiles = (N + 15) >> 4;
    const int RN = R * N;

    rgat_gemm_kernel<<<(R + 1) * numTiles, 128, 0, stream>>>(
        x, W, W_self, att_src, att_dst, H, S, Dt, N, R);

    rgat_selfmax_kernel<<<(RN + 255) / 256, 256, 0, stream>>>(S, Dt, Menc, RN);

    rgat_edgemax_kernel<<<(E + 255) / 256, 256, 0, stream>>>(
        src, dst, etype, S, Dt, Menc, N, E);

    rgat_deninit_kernel<<<(RN + 255) / 256, 256, 0, stream>>>(S, Dt, Menc, Den, RN);

    rgat_edgeden_kernel<<<(E + 255) / 256, 256, 0, stream>>>(
        src, dst, etype, S, Dt, Menc, Den, N, E);

    const size_t nf = (size_t)N * FF;
    rgat_baseout_kernel<<<(unsigned)((nf + 255) / 256), 256, 0, stream>>>(
        H, S, Dt, Menc, Den, b_self, bias, out, N, R);

    rgat_scatter_kernel<<<(E + 7) / 8, 256, 0, stream>>>(
        src, dst, etype, S, Dt, Menc, Den, H, out, N, E);
}